// PyTorchOrderParameters_20298015441444
// MI455X (gfx1250) — compile-verified
//
#include <hip/hip_runtime.h>
#include <hip/hip_bf16.h>

typedef __attribute__((ext_vector_type(2))) float v2f;
typedef __attribute__((ext_vector_type(8))) float v8f;

#define KNBR 64
#define WAVES_PER_BLOCK 8

__launch_bounds__(256)
__global__ void op_kernel(const float* __restrict__ pos,
                          const float* __restrict__ cell,
                          const int* __restrict__ atom_idx,
                          const int* __restrict__ nbr_idx,
                          const unsigned char* __restrict__ valid,
                          float* __restrict__ out,
                          int M)
{
    __shared__ float svx[WAVES_PER_BLOCK][KNBR];
    __shared__ float svy[WAVES_PER_BLOCK][KNBR];
    __shared__ float svz[WAVES_PER_BLOCK][KNBR];
    __shared__ float sva[WAVES_PER_BLOCK][KNBR];

    const int lane    = threadIdx.x & 31;
    const int wslot   = threadIdx.x >> 5;
    const int wglobal = (int)((blockIdx.x * blockDim.x + threadIdx.x) >> 5);
    const int m       = (wglobal < M) ? wglobal : (M - 1);

    // ---- cell and its inverse (uniform across all threads) ----
    const float c00 = cell[0], c01 = cell[1], c02 = cell[2];
    const float c10 = cell[3], c11 = cell[4], c12 = cell[5];
    const float c20 = cell[6], c21 = cell[7], c22 = cell[8];
    const float cofA = c11 * c22 - c12 * c21;
    const float cofB = c12 * c20 - c10 * c22;
    const float cofC = c10 * c21 - c11 * c20;
    const float invdet = 1.0f / (c00 * cofA + c01 * cofB + c02 * cofC);
    const float i00 = cofA * invdet;
    const float i01 = (c02 * c21 - c01 * c22) * invdet;
    const float i02 = (c01 * c12 - c02 * c11) * invdet;
    const float i10 = cofB * invdet;
    const float i11 = (c00 * c22 - c02 * c20) * invdet;
    const float i12 = (c02 * c10 - c00 * c12) * invdet;
    const float i20 = cofC * invdet;
    const float i21 = (c01 * c20 - c00 * c21) * invdet;
    const float i22 = (c00 * c11 - c01 * c10) * invdet;

    // ---- geometry: 2 neighbors per lane -> LDS ----
    const int center = atom_idx[m];
    const float cx = pos[3 * center + 0];
    const float cy = pos[3 * center + 1];
    const float cz = pos[3 * center + 2];

    float cnpart = 0.0f;
#pragma unroll
    for (int t = 0; t < 2; ++t) {
        const int j  = lane + 32 * t;
        const int ni = nbr_idx[m * KNBR + j];
        const float dx = pos[3 * ni + 0] - cx;
        const float dy = pos[3 * ni + 1] - cy;
        const float dz = pos[3 * ni + 2] - cz;
        // frac = dr @ cell_inv ; minimum image (rintf == round-half-to-even == jnp.round)
        float f0 = dx * i00 + dy * i10 + dz * i20;
        float f1 = dx * i01 + dy * i11 + dz * i21;
        float f2 = dx * i02 + dy * i12 + dz * i22;
        f0 -= rintf(f0); f1 -= rintf(f1); f2 -= rintf(f2);
        const float px = f0 * c00 + f1 * c10 + f2 * c20;
        const float py = f0 * c01 + f1 * c11 + f2 * c21;
        const float pz = f0 * c02 + f1 * c12 + f2 * c22;
        const float dist = sqrtf(px * px + py * py + pz * pz + 1e-10f);
        const float inv  = 1.0f / (dist + 1e-10f);
        const float vm   = (valid[m * KNBR + j] != 0) ? 1.0f : 0.0f;
        svx[wslot][j] = px * inv * vm;
        svy[wslot][j] = py * inv * vm;
        svz[wslot][j] = pz * inv * vm;
        sva[wslot][j] = vm;
        cnpart += vm;
    }
    __syncthreads();

    // ---- build WMMA fragments: 32-bit 16x4 A layout ----
    // lanes 0-15 (M=lane): VGPR0=K0(vx), VGPR1=K1(vy)
    // lanes 16-31        : VGPR0=K2(vz), VGPR1=K3(0)
    // B (4x16) has identical per-lane content for the symmetric Gram matrix.
    const int half = lane >> 4;
    const int l15  = lane & 15;
    v2f frag[4];
#pragma unroll
    for (int b = 0; b < 4; ++b) {
        const int idx = b * 16 + l15;
        const float ax = svx[wslot][idx];
        const float ay = svy[wslot][idx];
        const float az = svz[wslot][idx];
        frag[b][0] = (half == 0) ? ax : az;
        frag[b][1] = (half == 0) ? ay : 0.0f;
    }

    // ---- constants (divisions pre-folded into Gaussian coefficients) ----
    const float PI_F    = 3.14159265358979323846f;
    const float HPI_F   = 1.57079632679489661923f;
    const float TET_TA  = 1.9106332362490186f;      // 0.6081734479693927 * pi
    const float OCT_THR = 2.792526803190927f;
    // K = -0.5 / sigma^2  (tet sigma has the reference's +1e-10)
    const float K_TET = -0.5f / ((0.20943951023931953f + 1e-10f) *
                                 (0.20943951023931953f + 1e-10f));
    const float K_180 = -0.5f / (0.20943951023931953f * 0.20943951023931953f);
    const float K_90  = -0.5f / (0.17453292519943295f * 0.17453292519943295f);
    // NOTE: the reference's tet dot-clip to +-(1-1e-7) and exponent clip at -50
    // are numerically dead here: angle in [0,pi] bounds the exponent to
    // [-41.61, 0] > -50, and the clipped-vs-unclipped angle changes terms of
    // magnitude ~1e-18 only. Both clamps are omitted.

    float tet_sum = 0.0f, oct_sum = 0.0f;

    // Symmetric Gram matrix: process only upper-triangular tiles,
    // off-diagonal tiles contribute twice.
#pragma unroll
    for (int rb = 0; rb < 4; ++rb) {
        float validM[8];
#pragma unroll
        for (int r = 0; r < 8; ++r)
            validM[r] = sva[wslot][rb * 16 + r + 8 * half];

#pragma unroll
        for (int cb = 0; cb < 4; ++cb) {
            if (cb < rb) continue;                    // symmetry: skip lower tiles
            const float w = (cb == rb) ? 1.0f : 2.0f; // off-diag counts twice

            v8f acc = {0.f, 0.f, 0.f, 0.f, 0.f, 0.f, 0.f, 0.f};
            // D = A(16x4) x B(4x16) + C : full 3-D dot in one WMMA (4th K coord = 0)
            acc = __builtin_amdgcn_wmma_f32_16x16x4_f32(
                false, frag[rb], false, frag[cb], (short)0, acc, false, false);

            const int   nIdx   = cb * 16 + l15;
            const float validN = sva[wslot][nIdx] * w;
#pragma unroll
            for (int r = 0; r < 8; ++r) {
                const int mIdx = rb * 16 + r + 8 * half;
                // k==j exclusion only possible on diagonal tiles (disjoint
                // index ranges otherwise) -- rb/cb are compile-time constants.
                float fm = validM[r] * validN;
                if (rb == cb) fm *= (mIdx != nIdx) ? 1.0f : 0.0f;
                const float dot = acc[r];

                // clamp dot to [-1,1] before acos
                const float doo = fminf(fmaxf(dot, -1.0f), 1.0f);
                const float ao  = acosf(doo);

                // tetrahedral Gaussian: exp(K_TET*(ao-TA)^2)
                const float dt = ao - TET_TA;
                tet_sum = fmaf(__expf(dt * dt * K_TET), fm, tet_sum);

                // octahedral: select branch first, single exp
                const bool  south = (ao >= OCT_THR);
                const float co  = south ? PI_F  : HPI_F;
                const float ko  = south ? K_180 : K_90;
                const float amp = south ? 3.0f  : 1.0f;
                const float d2  = ao - co;
                const float go  = __expf(d2 * d2 * ko);
                oct_sum = fmaf(go, amp * fm, oct_sum);
            }
        }
    }

    // ---- wave32 reduction ----
#pragma unroll
    for (int off = 16; off > 0; off >>= 1) {
        tet_sum += __shfl_xor(tet_sum, off, 32);
        oct_sum += __shfl_xor(oct_sum, off, 32);
        cnpart  += __shfl_xor(cnpart,  off, 32);
    }

    if (lane == 0 && wglobal < M) {
        const float cn    = cnpart;
        const float n1    = fmaxf(cn, 1.0f);
        const float normt = n1 * fmaxf(n1 - 1.0f, 1.0f);
        const float tet   = tet_sum / (normt + 1e-10f);
        const float normo = cn * (3.0f + (cn - 2.0f) * (cn - 3.0f));
        const float octv  = oct_sum / (normo + 1e-10f);
        out[m]         = cn;
        out[M + m]     = tet;
        out[2 * M + m] = octv;
    }
}

extern "C" void kernel_launch(void* const* d_in, const int* in_sizes, int n_in,
                              void* d_out, int out_size, void* d_ws, size_t ws_size,
                              hipStream_t stream) {
    const float*         pos  = (const float*)d_in[0];
    const float*         cell = (const float*)d_in[1];
    const int*           aidx = (const int*)d_in[2];
    const int*           nidx = (const int*)d_in[3];
    const unsigned char* vmsk = (const unsigned char*)d_in[4];
    float*               out  = (float*)d_out;

    const int M = in_sizes[2];            // number of center atoms
    const int threads = 32 * WAVES_PER_BLOCK;
    const int blocks  = (M + WAVES_PER_BLOCK - 1) / WAVES_PER_BLOCK;

    op_kernel<<<blocks, threads, 0, stream>>>(pos, cell, aidx, nidx, vmsk, out, M);
}